// ModelEmbeddings_2551210574270
// MI455X (gfx1250) — compile-verified
//
#include <hip/hip_runtime.h>
#include <math.h>

typedef float v2f __attribute__((ext_vector_type(2)));
typedef float v8f __attribute__((ext_vector_type(8)));

#define B_     128
#define S_     256
#define MW_    21
#define ECHAR  50
#define EWORD  256
#define KW     5
#define NWORDS (B_ * S_)        // 32768
#define NPOS   (MW_ - KW + 1)   // 17
#define KIM    (ECHAR * KW)     // 250 (im2col K)
#define KPAD   252              // padded to multiple of 4
#define NPAIR  (KPAD / 2)       // 126 K-pairs (conv)

// Pair-interleaved layouts: element (q, o) of a [K][N] matrix lives at
//   W[(q>>1)*2*N + o*2 + (q&1)]
// so the {q, q+1} pair a lane needs for a WMMA fragment is one aligned
// 8-byte load (ds_load_b64 / global_load_b64), consumed directly as v2f.

// workspace layout (in floats)
#define WS_WT  0                               // WtI[126][256][2]  = 64512
#define WS_WP  (WS_WT + NPAIR * 2 * EWORD)     // WpI[128][256][2]  = 65536
#define WS_WG  (WS_WP + EWORD * EWORD)         // WgI[128][256][2]
#define WS_H   (WS_WG + EWORD * EWORD)         // H[32768][256]

#define SA_PAIRS   526   // (21*50 rows + 2 zero-pad) / 2
#define SA_STRIDE  32    // 16 words * 2; 32 % 64 == 32 -> conflict-free phases
#define SB_PAIRS   126
#define SB_STRIDE  512   // 256 cols * 2; bank-fixed via hf XOR-16 column swizzle
#define SH_PAIRS   128
#define SH_STRIDE  96    // 32 words * 2 + pad; 96 % 64 == 32 -> conflict-free

__device__ __forceinline__ v8f wmma_f32(v2f a, v2f b, v8f c) {
  // D = A(16x4,f32) * B(4x16,f32) + C(16x16,f32)
  return __builtin_amdgcn_wmma_f32_16x16x4_f32(false, a, false, b, (short)0, c,
                                               false, false);
}

// ---------------------------------------------------------------------------
// Kernel 0: build pair-interleaved, transposed weights in workspace.
//   WtI: conv_w[o][c][k] -> (q = k*50+c, o) pairs, q >= 250 zero-padded
//   WpI/WgI: w_proj/w_gate[o][c] -> (c, o) pairs
// ---------------------------------------------------------------------------
__global__ void __launch_bounds__(256)
prep_kernel(const float* __restrict__ conv_w, const float* __restrict__ w_proj,
            const float* __restrict__ w_gate, float* __restrict__ ws) {
  int i = blockIdx.x * 256 + threadIdx.x;
  const int n1 = NPAIR * 2 * EWORD;   // 64512
  const int n2 = EWORD * EWORD;       // 65536
  if (i < n1) {
    int pair = i >> 9, e = i & 511;
    int o = e >> 1, par = e & 1;
    int q = pair * 2 + par;
    float v = 0.0f;
    if (q < KIM) {
      int k = q / ECHAR, c = q % ECHAR;
      v = conv_w[(o * ECHAR + c) * KW + k];
    }
    ws[WS_WT + i] = v;
  } else if (i < n1 + 2 * n2) {
    int j = i - n1;
    int m = (j < n2) ? 0 : 1;          // 0 -> proj, 1 -> gate
    int jj = j - m * n2;
    int pair = jj >> 9, e = jj & 511;
    int o = e >> 1, par = e & 1;
    int c = pair * 2 + par;
    const float* w = m ? w_gate : w_proj;
    ws[(m ? WS_WG : WS_WP) + jj] = w[o * EWORD + c];
  }
}

// ---------------------------------------------------------------------------
// Kernel 1: embedding gather + conv1d (im2col WMMA GEMM) + bias + ReLU +
// maxpool over 17 positions. Workgroup: 16 words x 256 channels, 8 wave32s,
// each wave owns 2 N-tiles. K-step = 1x ds_load_b64 (A) + 2x ds_load_b64 (B)
// -> 2 WMMAs. All loop addresses are incremented base pointers + compile-time
// immediate offsets (K-loop = 7 blocks x 9 unrolled steps).
// ---------------------------------------------------------------------------
__global__ void __launch_bounds__(256)
conv_kernel(const int* __restrict__ sents, const float* __restrict__ emb,
            const float* __restrict__ conv_b, float* __restrict__ ws) {
  extern __shared__ float smem[];
  float* sA = smem;                          // [526][16][2] row-pair interleave
  float* sB = smem + SA_PAIRS * SA_STRIDE;   // [126][256][2], hf-swizzled cols

  const float* WtI = ws + WS_WT;
  float* H = ws + WS_H;

  const int tid = threadIdx.x;
  const int wg = blockIdx.x;               // word tile (16 words)

  // Stage embeddings interleaved: (row=p*50+c, word) ->
  //   sA[(row>>1)*32 + word*2 + (row&1)]
  // (emb row 0 is zeroed in setup, so pad chars gather zeros naturally)
  for (int j = tid; j < 1050 * 16; j += 256) {
    int word = j & 15;
    int row = j >> 4;                 // row = p*50 + c
    int p = row / ECHAR;
    int c = row - p * ECHAR;
    int ch = sents[(wg * 16 + word) * MW_ + p];
    sA[(row >> 1) * SA_STRIDE + word * 2 + (row & 1)] = emb[ch * ECHAR + c];
  }
  if (tid < 32) sA[(SA_PAIRS - 1) * SA_STRIDE + tid] = 0.0f;  // rows 1050,1051

  // Stage full interleaved weight slab with hf bank swizzle:
  // column stored at col ^ (16 * pair_parity)
  for (int j = tid; j < SB_PAIRS * 512; j += 256) {
    int pair = j >> 9;
    int e = j & 511;
    int col = e >> 1, par = e & 1;
    int colx = col ^ ((pair & 1) << 4);
    sB[pair * SB_STRIDE + colx * 2 + par] = WtI[j];
  }
  __syncthreads();

  const int wave = tid >> 5;
  const int lane = tid & 31;
  const int hf = lane >> 4;   // K-phase half: lanes 16-31 handle K+2,K+3
  const int lm = lane & 15;
  const int c0 = wave * 32;   // 2 N-tiles per wave

  // pairB = kk/2 + hf has parity hf (kk/2 always even) -> swizzle is a
  // per-lane constant folded into the base pointers.
  const float* bbase0 = sB + hf * SB_STRIDE + ((c0 + lm) ^ (hf << 4)) * 2;
  const float* bbase1 = sB + hf * SB_STRIDE + ((c0 + 16 + lm) ^ (hf << 4)) * 2;

  const float bias0 = conv_b[c0 + lm];
  const float bias1 = conv_b[c0 + 16 + lm];

  v8f mx0 = (v8f)0.0f, mx1 = (v8f)0.0f;  // running max(ReLU(y_t))

  for (int t = 0; t < NPOS; ++t) {
    v8f acc0 = (v8f)0.0f, acc1 = (v8f)0.0f;
    // A row-pair base for this t: (t*50 + kk + 2*hf) >> 1, advancing 2/step
    const float* ap = sA + (t * 25 + hf) * SA_STRIDE + lm * 2;
    const float* bp0 = bbase0;
    const float* bp1 = bbase1;
    for (int blk = 0; blk < 7; ++blk) {   // 63 K-steps = 7 x 9
#pragma unroll
      for (int u = 0; u < 9; ++u) {
        v2f a = *(const v2f*)(ap + u * (2 * SA_STRIDE));
        v2f b0 = *(const v2f*)(bp0 + u * (2 * SB_STRIDE));
        v2f b1 = *(const v2f*)(bp1 + u * (2 * SB_STRIDE));
        acc0 = wmma_f32(a, b0, acc0);
        acc1 = wmma_f32(a, b1, acc1);
      }
      ap += 9 * (2 * SA_STRIDE);
      bp0 += 9 * (2 * SB_STRIDE);
      bp1 += 9 * (2 * SB_STRIDE);
    }
    for (int r = 0; r < 8; ++r) {
      mx0[r] = fmaxf(mx0[r], fmaxf(acc0[r] + bias0, 0.0f));
      mx1[r] = fmaxf(mx1[r], fmaxf(acc1[r] + bias1, 0.0f));
    }
  }

  // C/D layout: VGPR r, lanes 0-15 -> word r, lanes 16-31 -> word r+8
  for (int r = 0; r < 8; ++r) {
    int word = wg * 16 + r + 8 * hf;
    H[word * EWORD + c0 + lm] = mx0[r];
    H[word * EWORD + c0 + 16 + lm] = mx1[r];
  }
}

// ---------------------------------------------------------------------------
// Kernel 2: highway network. Both 256x256 GEMMs (proj & gate) accumulated
// simultaneously. Workgroup: 32 words x 256 channels; 8 waves x 2 N-tiles
// x 2 M-tiles -> 8 WMMAs per K-step vs 2 ds_load_b64 + 4 global b64.
// K-loop = 8 blocks x 8 unrolled steps with immediate offsets.
// ---------------------------------------------------------------------------
__global__ void __launch_bounds__(256)
highway_kernel(const float* __restrict__ b_proj, const float* __restrict__ b_gate,
               const float* __restrict__ ws_c, float* __restrict__ out) {
  __shared__ float sH[SH_PAIRS * SH_STRIDE];  // [128 c-pairs][32 words][2]+pad
  const float* WpI = ws_c + WS_WP;
  const float* WgI = ws_c + WS_WG;
  const float* H = ws_c + WS_H;

  const int tid = threadIdx.x;
  const int wg = blockIdx.x;

  for (int w = 0; w < 32; ++w)  // coalesced global reads of H tile
    sH[(tid >> 1) * SH_STRIDE + w * 2 + (tid & 1)] = H[(wg * 32 + w) * EWORD + tid];
  __syncthreads();

  const int wave = tid >> 5;
  const int lane = tid & 31;
  const int hf = lane >> 4;
  const int lm = lane & 15;
  const int c0 = wave * 32;  // 2 N-tiles per wave

  const float bp0 = b_proj[c0 + lm], bp1 = b_proj[c0 + 16 + lm];
  const float bg0 = b_gate[c0 + lm], bg1 = b_gate[c0 + 16 + lm];

  v8f aP00 = (v8f)0.0f, aP01 = (v8f)0.0f, aP10 = (v8f)0.0f, aP11 = (v8f)0.0f;
  v8f aG00 = (v8f)0.0f, aG01 = (v8f)0.0f, aG10 = (v8f)0.0f, aG11 = (v8f)0.0f;

  const float* hp = sH + hf * SH_STRIDE + lm * 2;
  const float* p0 = WpI + hf * 512 + (c0 + lm) * 2;
  const float* p1 = WpI + hf * 512 + (c0 + 16 + lm) * 2;
  const float* g0 = WgI + hf * 512 + (c0 + lm) * 2;
  const float* g1 = WgI + hf * 512 + (c0 + 16 + lm) * 2;

  for (int blk = 0; blk < 8; ++blk) {  // 64 K-steps = 8 x 8
#pragma unroll
    for (int u = 0; u < 8; ++u) {
      v2f a0 = *(const v2f*)(hp + u * (2 * SH_STRIDE));
      v2f a1 = *(const v2f*)(hp + u * (2 * SH_STRIDE) + 32);
      v2f bP0 = *(const v2f*)(p0 + u * 1024);
      v2f bP1 = *(const v2f*)(p1 + u * 1024);
      v2f bG0 = *(const v2f*)(g0 + u * 1024);
      v2f bG1 = *(const v2f*)(g1 + u * 1024);
      aP00 = wmma_f32(a0, bP0, aP00);
      aP01 = wmma_f32(a0, bP1, aP01);
      aP10 = wmma_f32(a1, bP0, aP10);
      aP11 = wmma_f32(a1, bP1, aP11);
      aG00 = wmma_f32(a0, bG0, aG00);
      aG01 = wmma_f32(a0, bG1, aG01);
      aG10 = wmma_f32(a1, bG0, aG10);
      aG11 = wmma_f32(a1, bG1, aG11);
    }
    hp += 8 * (2 * SH_STRIDE);
    p0 += 8 * 1024;
    p1 += 8 * 1024;
    g0 += 8 * 1024;
    g1 += 8 * 1024;
  }

  for (int mt = 0; mt < 2; ++mt) {
    const v8f& aP_0 = mt ? aP10 : aP00;
    const v8f& aP_1 = mt ? aP11 : aP01;
    const v8f& aG_0 = mt ? aG10 : aG00;
    const v8f& aG_1 = mt ? aG11 : aG01;
    for (int r = 0; r < 8; ++r) {
      int word = mt * 16 + r + 8 * hf;
      int gword = wg * 32 + word;
      {
        int col = c0 + lm;
        float y = sH[(col >> 1) * SH_STRIDE + word * 2 + (col & 1)];
        float p = fmaxf(aP_0[r] + bp0, 0.0f);
        float g = 1.0f / (1.0f + __expf(-(aG_0[r] + bg0)));
        out[gword * EWORD + col] = g * p + (1.0f - g) * y;
      }
      {
        int col = c0 + 16 + lm;
        float y = sH[(col >> 1) * SH_STRIDE + word * 2 + (col & 1)];
        float p = fmaxf(aP_1[r] + bp1, 0.0f);
        float g = 1.0f / (1.0f + __expf(-(aG_1[r] + bg1)));
        out[gword * EWORD + col] = g * p + (1.0f - g) * y;
      }
    }
  }
}

// ---------------------------------------------------------------------------
extern "C" void kernel_launch(void* const* d_in, const int* in_sizes, int n_in,
                              void* d_out, int out_size, void* d_ws, size_t ws_size,
                              hipStream_t stream) {
  (void)in_sizes; (void)n_in; (void)out_size; (void)ws_size;
  const int* sents = (const int*)d_in[0];
  const float* emb = (const float*)d_in[1];
  const float* conv_w = (const float*)d_in[2];
  const float* conv_b = (const float*)d_in[3];
  const float* w_proj = (const float*)d_in[4];
  const float* b_proj = (const float*)d_in[5];
  const float* w_gate = (const float*)d_in[6];
  const float* b_gate = (const float*)d_in[7];
  float* ws = (float*)d_ws;
  float* out = (float*)d_out;

  const int prep_n = NPAIR * 2 * EWORD + 2 * EWORD * EWORD;
  prep_kernel<<<(prep_n + 255) / 256, 256, 0, stream>>>(conv_w, w_proj, w_gate, ws);

  const size_t conv_lds =
      (size_t)(SA_PAIRS * SA_STRIDE + SB_PAIRS * SB_STRIDE) * sizeof(float);
  conv_kernel<<<NWORDS / 16, 256, conv_lds, stream>>>(sents, emb, conv_b, ws);

  highway_kernel<<<NWORDS / 32, 256, 0, stream>>>(b_proj, b_gate, ws, out);
}